// GCN_63582695850896
// MI455X (gfx1250) — compile-verified
//
#include <hip/hip_runtime.h>
#include <hip/hip_bf16.h>

typedef __attribute__((ext_vector_type(2))) float v2f;
typedef __attribute__((ext_vector_type(8))) float v8f;

#define HGCN 128

// ---------------------------------------------------------------------------
// Degree / normalization setup
// ---------------------------------------------------------------------------
__global__ void GCN_deg_init(float* deg, int n) {
    int i = blockIdx.x * blockDim.x + threadIdx.x;
    if (i < n) deg[i] = 1.0f;  // self-loop contributes 1
}

__global__ void GCN_deg_count(const int* __restrict__ rows, float* deg, int E) {
    int e = blockIdx.x * blockDim.x + threadIdx.x;
    if (e < E) atomicAdd(&deg[rows[e]], 1.0f);
}

__global__ void GCN_dinv(const float* __restrict__ deg, float* dinv, int n) {
    int i = blockIdx.x * blockDim.x + threadIdx.x;
    if (i < n) {
        float d = deg[i];
        dinv[i] = (d > 0.0f) ? rsqrtf(d) : 0.0f;
    }
}

__global__ void GCN_val(const int* __restrict__ rows, const int* __restrict__ cols,
                        const float* __restrict__ dinv, float* val, int E) {
    int e = blockIdx.x * blockDim.x + threadIdx.x;
    if (e < E) val[e] = dinv[rows[e]] * dinv[cols[e]];
}

// ---------------------------------------------------------------------------
// Elementwise helpers
// ---------------------------------------------------------------------------
__global__ void GCN_zero(float* p, long long n) {
    long long i = (long long)blockIdx.x * blockDim.x + threadIdx.x;
    if (i < n) p[i] = 0.0f;
}

__global__ void GCN_copy(float* __restrict__ dst, const float* __restrict__ src, long long n) {
    long long i = (long long)blockIdx.x * blockDim.x + threadIdx.x;
    if (i < n) dst[i] = src[i];
}

// ---------------------------------------------------------------------------
// SpMM scatter: out[row] += scale * val * h[col]   (one wave per edge,
// each lane handles 4 of the 128 features; atomics land in L2 since the
// 25.6MB feature matrix is L2-resident on MI455X's 192MB L2)
// ---------------------------------------------------------------------------
__global__ void __launch_bounds__(256)
GCN_spmm(const float* __restrict__ h, float* __restrict__ out,
         const int* __restrict__ rows, const int* __restrict__ cols,
         const float* __restrict__ val, const float* __restrict__ dinv,
         int E, int n, float scale) {
    long long g = (long long)blockIdx.x * blockDim.x + threadIdx.x;
    int e = (int)(g >> 5);
    int lane = (int)(g & 31);
    if (e >= E + n) return;
    int r, c; float v;
    if (e < E) {
        r = rows[e]; c = cols[e]; v = val[e];
    } else {
        int i = e - E;
        r = i; c = i;
        float d = dinv[i];
        v = d * d;
    }
    v *= scale;
    const float4 hv = *(const float4*)(h + (size_t)c * HGCN + lane * 4);
    float* dst = out + (size_t)r * HGCN + lane * 4;
    atomicAdd(dst + 0, v * hv.x);
    atomicAdd(dst + 1, v * hv.y);
    atomicAdd(dst + 2, v * hv.z);
    atomicAdd(dst + 3, v * hv.w);
}

// ---------------------------------------------------------------------------
// WMMA fp32 GEMM:  out[N x MCOLS] (+)= A[N x 128] @ W[128 x MCOLS] + bias
// Block = 128 threads = 4 waves; each wave computes a 16-row strip across the
// full output width using V_WMMA_F32_16X16X4_F32 (MCOLS/16 accumulators).
// W is staged in LDS pair-interleaved ([k/2][col][k&1]) so each B fragment is
// one ds_load_b64; each A fragment is one 8-byte global load.
//
// f32 WMMA fragment layout (wave32): lanes 0-15 hold K={0,1} (v0,v1),
// lanes 16-31 hold K={2,3}; A row / B col index = lane & 15.
// ---------------------------------------------------------------------------
template <int MCOLS, int ACC>
__global__ void __launch_bounds__(128)
GCN_gemm_wmma(const float* __restrict__ A, const float* __restrict__ W,
              const float* __restrict__ bias, float* __restrict__ out, int n) {
    constexpr int K = 128;
    constexpr int MT = MCOLS / 16;
    __shared__ float wl[K * MCOLS];

    const int tid = threadIdx.x;
    // cooperative pair-interleaved load of W into LDS
    for (int i = tid; i < K * MCOLS; i += 128) {
        int k = i / MCOLS;
        int c = i - k * MCOLS;
        wl[(k >> 1) * (2 * MCOLS) + c * 2 + (k & 1)] = W[i];
    }
    __syncthreads();

    const int wave  = tid >> 5;
    const int lane  = tid & 31;
    const int m     = lane & 15;        // A row / B col within tile
    const int khalf = (lane >> 4) * 2;  // 0 or 2

    const int row0 = blockIdx.x * 64 + wave * 16;
    int arow = row0 + m;
    if (arow > n - 1) arow = n - 1;     // clamp loads; stores are guarded
    const float* aptr = A + (size_t)arow * K + khalf;

    v8f acc[MT] = {};

    for (int kb = 0; kb < K; kb += 4) {
        const int k = kb + khalf;
        v2f a = *(const v2f*)(aptr + kb);
        const float* bbase = wl + (k >> 1) * (2 * MCOLS) + m * 2;
#pragma unroll
        for (int t = 0; t < MT; ++t) {
            v2f b = *(const v2f*)(bbase + t * 32);
            acc[t] = __builtin_amdgcn_wmma_f32_16x16x4_f32(
                false, a, false, b, (short)0, acc[t], false, false);
        }
    }

    // C/D layout: VGPR r -> row (row0 + r) for lanes 0-15, (row0 + 8 + r) for 16-31
    const int srowbase = row0 + ((lane >> 4) << 3);
#pragma unroll
    for (int t = 0; t < MT; ++t) {
        int col = t * 16 + m;
        float bv = bias[col];
#pragma unroll
        for (int r = 0; r < 8; ++r) {
            int row = srowbase + r;
            if (row < n) {
                float v = acc[t][r] + bv;
                size_t idx = (size_t)row * MCOLS + col;
                if (ACC) out[idx] += v;
                else     out[idx] = v;
            }
        }
    }
}

// ---------------------------------------------------------------------------
// Fused LayerNorm + ReLU + residual:  out = relu(LN(u)*g + beta) + s
// One wave per row; 4 features per lane; wave32 shfl_xor reductions.
// ---------------------------------------------------------------------------
__global__ void __launch_bounds__(256)
GCN_ln_relu_add(const float* __restrict__ u, const float* __restrict__ s,
                const float* __restrict__ g, const float* __restrict__ beta,
                float* __restrict__ out, int n) {
    int row  = blockIdx.x * (blockDim.x >> 5) + (threadIdx.x >> 5);
    int lane = threadIdx.x & 31;
    if (row >= n) return;
    size_t base = (size_t)row * HGCN + lane * 4;
    float4 uv = *(const float4*)(u + base);

    float sum = uv.x + uv.y + uv.z + uv.w;
#pragma unroll
    for (int off = 16; off; off >>= 1) sum += __shfl_xor(sum, off, 32);
    float mu = sum * (1.0f / HGCN);

    float dx = uv.x - mu, dy = uv.y - mu, dz = uv.z - mu, dw = uv.w - mu;
    float vs = dx * dx + dy * dy + dz * dz + dw * dw;
#pragma unroll
    for (int off = 16; off; off >>= 1) vs += __shfl_xor(vs, off, 32);
    float rstd = rsqrtf(vs * (1.0f / HGCN) + 1e-5f);

    float4 gv = *(const float4*)(g + lane * 4);
    float4 bv = *(const float4*)(beta + lane * 4);
    float4 sv = *(const float4*)(s + base);

    float4 o;
    o.x = fmaxf(dx * rstd * gv.x + bv.x, 0.0f) + sv.x;
    o.y = fmaxf(dy * rstd * gv.y + bv.y, 0.0f) + sv.y;
    o.z = fmaxf(dz * rstd * gv.z + bv.z, 0.0f) + sv.z;
    o.w = fmaxf(dw * rstd * gv.w + bv.w, 0.0f) + sv.w;
    *(float4*)(out + base) = o;
}

// ---------------------------------------------------------------------------
extern "C" void kernel_launch(void* const* d_in, const int* in_sizes, int n_in,
                              void* d_out, int out_size, void* d_ws, size_t ws_size,
                              hipStream_t stream) {
    const float* x    = (const float*)d_in[0];
    const int*   ei   = (const int*)d_in[1];
    const float* w_in = (const float*)d_in[2];
    const float* b_in = (const float*)d_in[3];
    const float* W1[2] = {(const float*)d_in[4],  (const float*)d_in[10]};
    const float* Bb[2] = {(const float*)d_in[5],  (const float*)d_in[11]};
    const float* Gg[2] = {(const float*)d_in[6],  (const float*)d_in[12]};
    const float* Bt[2] = {(const float*)d_in[7],  (const float*)d_in[13]};
    const float* Ws[2] = {(const float*)d_in[8],  (const float*)d_in[14]};
    const float* Bs[2] = {(const float*)d_in[9],  (const float*)d_in[15]};
    const float* w3  = (const float*)d_in[16];
    const float* b3  = (const float*)d_in[17];
    const float* ws3 = (const float*)d_in[18];
    const float* bs3 = (const float*)d_in[19];

    const int N = in_sizes[0] / HGCN;
    const int E = in_sizes[1] / 2;
    const int* rows = ei;
    const int* cols = ei + E;

    char* ws = (char*)d_ws;
    const size_t NB = (size_t)N * HGCN * sizeof(float);
    float* B0   = (float*)(ws);            // h_in
    float* B1   = (float*)(ws + 1 * NB);   // t1 = A*h   (later: skip branch)
    float* B2   = (float*)(ws + 2 * NB);   // cheb
    float* B3   = (float*)(ws + 3 * NB);   // u = cheb@W + b
    float* deg  = (float*)(ws + 4 * NB);
    float* dinv = deg + N;
    float* val  = dinv + N;

    const long long nelem = (long long)N * HGCN;
    const int zb = (int)((nelem + 255) / 256);
    const long long spthreads = (long long)(E + N) * 32;
    const int spb = (int)((spthreads + 255) / 256);
    const dim3 gg((N + 63) / 64);

    // --- normalization coefficients (recomputed every call) ---
    GCN_deg_init<<<(N + 255) / 256, 256, 0, stream>>>(deg, N);
    GCN_deg_count<<<(E + 255) / 256, 256, 0, stream>>>(rows, deg, E);
    GCN_dinv<<<(N + 255) / 256, 256, 0, stream>>>(deg, dinv, N);
    GCN_val<<<(E + 255) / 256, 256, 0, stream>>>(rows, cols, dinv, val, E);

    // --- input projection: B0 = x @ w_in + b_in ---
    GCN_gemm_wmma<128, 0><<<gg, 128, 0, stream>>>(x, w_in, b_in, B0, N);

    // --- two hidden layers ---
    for (int l = 0; l < 2; ++l) {
        GCN_zero<<<zb, 256, 0, stream>>>(B1, nelem);
        GCN_spmm<<<spb, 256, 0, stream>>>(B0, B1, rows, cols, val, dinv, E, N, 1.0f);
        GCN_copy<<<zb, 256, 0, stream>>>(B2, B1, nelem);
        GCN_spmm<<<spb, 256, 0, stream>>>(B1, B2, rows, cols, val, dinv, E, N, 2.0f);
        GCN_gemm_wmma<128, 0><<<gg, 128, 0, stream>>>(B2, W1[l], Bb[l], B3, N);
        GCN_gemm_wmma<128, 0><<<gg, 128, 0, stream>>>(B0, Ws[l], Bs[l], B1, N);
        GCN_ln_relu_add<<<(N + 7) / 8, 256, 0, stream>>>(B3, B1, Gg[l], Bt[l], B0, N);
    }

    // --- output layer (Identity norm, no activation), DOUT = 64 ---
    GCN_zero<<<zb, 256, 0, stream>>>(B1, nelem);
    GCN_spmm<<<spb, 256, 0, stream>>>(B0, B1, rows, cols, val, dinv, E, N, 1.0f);
    GCN_copy<<<zb, 256, 0, stream>>>(B2, B1, nelem);
    GCN_spmm<<<spb, 256, 0, stream>>>(B1, B2, rows, cols, val, dinv, E, N, 2.0f);
    GCN_gemm_wmma<64, 0><<<gg, 128, 0, stream>>>(B2, w3, b3, (float*)d_out, N);
    GCN_gemm_wmma<64, 1><<<gg, 128, 0, stream>>>(B0, ws3, bs3, (float*)d_out, N);
}